// InvariantPointAttention_75067438399489
// MI455X (gfx1250) — compile-verified
//
#include <hip/hip_runtime.h>
#include <hip/hip_bf16.h>
#include <math.h>

#define NRES 512
#define CS   384
#define CZ   128
#define CH   16
#define NH   12
#define PQ   4
#define PV   8
#define NB   64
#define ODIM 2112           // H * (CH + PV*3 + PV + CZ) = 12*176
#define NN   (NRES*NRES)

typedef __attribute__((ext_vector_type(2))) float v2f;
typedef __attribute__((ext_vector_type(8))) float v8f;

__device__ __forceinline__ v8f wmma4(v2f a, v2f b, v8f c) {
  // V_WMMA_F32_16X16X4_F32 : D = A(16x4) x B(4x16) + C, full fp32
  return __builtin_amdgcn_wmma_f32_16x16x4_f32(false, a, false, b, (short)0, c,
                                               false, false);
}

// ---------------------------------------------------------------------------
// Generic fp32 WMMA GEMM: C[M x Ncols] = A[M x K] @ B[K x Ncols] (+ bias)
// 4 waves / block, each wave owns a 16x16 output tile (consecutive M tiles).
// M multiple of 64; K multiple of 4. FULL=true => Ncols multiple of 16
// (no column masking at all). FULL=false => clamped always-valid addresses +
// v_cndmask zeroing — still no exec branching in the inner loop.
// A fragment layout (ISA 7.12.2, 32-bit A 16x4): lanes 0-15 -> K 0,1 ;
// lanes 16-31 -> K 2,3 ; M = lane&15. B mirrored; C/D rows r + 8*(lane>>4).
// ---------------------------------------------------------------------------
template <bool FULL>
__global__ __launch_bounds__(128)
void gemm_f32(const float* __restrict__ A, int lda,
              const float* __restrict__ B, int ldb,
              const float* __restrict__ bias,
              float* __restrict__ C, int ldc,
              int Ncols, int K) {
  const int wave = threadIdx.x >> 5;
  const int lane = threadIdx.x & 31;
  const int hi = lane >> 4, l16 = lane & 15;
  const int tm = (blockIdx.x * 4 + wave) * 16;
  const int tn = blockIdx.y * 16;
  const int bcol = tn + l16;
  const bool cok = FULL ? true : (bcol < Ncols);
  const int bcs = FULL ? bcol : (cok ? bcol : 0);   // clamped, always valid
  const float* Arow = A + (long)(tm + l16) * lda + hi * 2;
  const float* Bcol = B + bcs + (long)(hi * 2) * ldb;
  v8f acc = {};
  for (int k = 0; k < K; k += 4) {
    v2f a, b;
    a.x = Arow[k];
    a.y = Arow[k + 1];
    const float bx = Bcol[(long)k * ldb];
    const float by = Bcol[(long)(k + 1) * ldb];
    b.x = FULL ? bx : (cok ? bx : 0.f);
    b.y = FULL ? by : (cok ? by : 0.f);
    acc = wmma4(a, b, acc);
  }
  if (cok) {
    const float bv = bias ? bias[bcol] : 0.f;
#pragma unroll
    for (int r = 0; r < 8; ++r)
      C[(long)(tm + r + hi * 8) * ldc + bcol] = acc[r] + bv;
  }
}

// ---------------------------------------------------------------------------
// pair_bias: z[(n*m) x CZ] @ Wb[CZ x NH] + bb, stored transposed into
// logits L[h, n, m]  (dominant z-streaming GEMM). 4 waves / block.
// ---------------------------------------------------------------------------
__global__ __launch_bounds__(128)
void pair_bias_kernel(const float* __restrict__ z, const float* __restrict__ Wb,
                      const float* __restrict__ bb, float* __restrict__ L) {
  const int wave = threadIdx.x >> 5;
  const int lane = threadIdx.x & 31;
  const int hi = lane >> 4, l16 = lane & 15;
  const long tm = ((long)blockIdx.x * 4 + wave) * 16;   // flat (n*NRES + m) tile
  const bool cok = l16 < NH;
  const int hcs = cok ? l16 : 0;                        // clamped Wb column
  const float* Arow = z + (tm + l16) * CZ + hi * 2;
  const float* Bcol = Wb + hcs + hi * 2 * NH;
  v8f acc = {};
#pragma unroll 4
  for (int k = 0; k < CZ; k += 4) {
    v2f a, b;
    a.x = Arow[k];
    a.y = Arow[k + 1];
    const float bx = Bcol[k * NH];
    const float by = Bcol[(k + 1) * NH];
    b.x = cok ? bx : 0.f;
    b.y = cok ? by : 0.f;
    acc = wmma4(a, b, acc);
  }
  if (cok) {
    const float bv = bb[l16];
#pragma unroll
    for (int r = 0; r < 8; ++r) {
      const long row = tm + r + hi * 8;
      L[(long)l16 * NN + row] = acc[r] + bv;
    }
  }
}

// local -> global frame, in place: p = R*p + t
__global__ __launch_bounds__(256)
void to_global_kernel(float* __restrict__ pts, const float* __restrict__ rot,
                      const float* __restrict__ trans, int npts) {
  const int idx = blockIdx.x * 256 + threadIdx.x;
  if (idx >= NRES * npts) return;
  const int n = idx / npts, p = idx - n * npts;
  const float* R = rot + n * 9;
  float* v = pts + (long)(n * npts + p) * 3;
  const float x = v[0], y = v[1], zz = v[2];
  v[0] = R[0] * x + R[1] * y + R[2] * zz + trans[n * 3 + 0];
  v[1] = R[3] * x + R[4] * y + R[5] * zz + trans[n * 3 + 1];
  v[2] = R[6] * x + R[7] * y + R[8] * zz + trans[n * 3 + 2];
}

// ---------------------------------------------------------------------------
// Logits: 12 waves / block, wave h handles a 16(n) x 16(m) tile for head h.
// scalar q.k via 4 WMMA steps + point / dist-bin / multiscale terms in VALU,
// accumulated onto pre-written pair_bias in L[h,n,m].
// ---------------------------------------------------------------------------
__global__ __launch_bounds__(384)
void logits_kernel(const float* __restrict__ q, const float* __restrict__ kb,
                   const float* __restrict__ qg, const float* __restrict__ kg,
                   const float* __restrict__ trans,
                   const float* __restrict__ emb, const float* __restrict__ sl,
                   const float* __restrict__ hw, float* __restrict__ L) {
  const int h = threadIdx.x >> 5;
  const int lane = threadIdx.x & 31;
  const int hi = lane >> 4, l16 = lane & 15;
  const int tn = blockIdx.y * 16;  // n tile
  const int tm = blockIdx.x * 16;  // m tile

  const float* qrow = q + (tn + l16) * (NH * CH) + h * CH + hi * 2;
  const float* krow = kb + (tm + l16) * (NH * CH) + h * CH + hi * 2;
  v8f acc = {};
#pragma unroll
  for (int k = 0; k < CH; k += 4) {
    v2f a, b;
    a.x = qrow[k];
    a.y = qrow[k + 1];
    b.x = krow[k];
    b.y = krow[k + 1];
    acc = wmma4(a, b, acc);
  }

  // per-head softmax over 3 scale logits
  const float s0 = sl[h], s1 = sl[NH + h], s2 = sl[2 * NH + h];
  const float mx3 = fmaxf(s0, fmaxf(s1, s2));
  const float e0 = expf(s0 - mx3), e1 = expf(s1 - mx3), e2 = expf(s2 - mx3);
  const float inv3 = 1.f / (e0 + e1 + e2);
  const float ph0 = e0 * inv3, ph1 = e1 * inv3, ph2 = e2 * inv3;
  const float hwh = hw[h];

  const int m = tm + l16;
  float kgp[PQ * 3];
#pragma unroll
  for (int j = 0; j < PQ * 3; ++j)
    kgp[j] = kg[m * (NH * PQ * 3) + h * (PQ * 3) + j];
  const float tmx = trans[m * 3 + 0], tmy = trans[m * 3 + 1],
              tmz = trans[m * 3 + 2];

#pragma unroll
  for (int r = 0; r < 8; ++r) {
    const int n = tn + r + hi * 8;
    const float* qgn = qg + n * (NH * PQ * 3) + h * (PQ * 3);
    float pl = 0.f;
#pragma unroll
    for (int p = 0; p < PQ; ++p) {
      const float dx = qgn[p * 3 + 0] - kgp[p * 3 + 0];
      const float dy = qgn[p * 3 + 1] - kgp[p * 3 + 1];
      const float dz = qgn[p * 3 + 2] - kgp[p * 3 + 2];
      pl += dx * dx + dy * dy + dz * dz;
    }
    pl *= -0.5f * hwh;
    const float ddx = trans[n * 3 + 0] - tmx;
    const float ddy = trans[n * 3 + 1] - tmy;
    const float ddz = trans[n * 3 + 2] - tmz;
    const float d = sqrtf(ddx * ddx + ddy * ddy + ddz * ddz);
    int bi = (int)ceilf(d * 2.f) - 1;                   // bin edges step = 0.5
    bi = bi < 0 ? 0 : (bi > NB - 1 ? NB - 1 : bi);
    const float db = emb[bi * NH + h];
    const float ms = (d <= 5.f ? ph0 : 0.f) +
                     ((d > 5.f && d <= 15.f) ? ph1 : 0.f) + ph2;
    const long idx = (long)h * NN + (long)n * NRES + m;
    L[idx] += acc[r] * 0.25f + pl + db + ms;            // 0.25 = 1/sqrt(CH)
  }
}

// softmax over contiguous m-rows of L[h,n,m]
__global__ __launch_bounds__(128)
void softmax_kernel(float* __restrict__ L) {
  __shared__ float red[128];
  float* p = L + (long)blockIdx.x * NRES;
  const int t = threadIdx.x;
  float v0 = p[t], v1 = p[t + 128], v2 = p[t + 256], v3 = p[t + 384];
  float mx = fmaxf(fmaxf(v0, v1), fmaxf(v2, v3));
  red[t] = mx; __syncthreads();
  for (int s = 64; s > 0; s >>= 1) {
    if (t < s) red[t] = fmaxf(red[t], red[t + s]);
    __syncthreads();
  }
  mx = red[0]; __syncthreads();
  v0 = expf(v0 - mx); v1 = expf(v1 - mx);
  v2 = expf(v2 - mx); v3 = expf(v3 - mx);
  red[t] = v0 + v1 + v2 + v3; __syncthreads();
  for (int s = 64; s > 0; s >>= 1) {
    if (t < s) red[t] += red[t + s];
    __syncthreads();
  }
  const float inv = 1.f / red[0];
  p[t] = v0 * inv; p[t + 128] = v1 * inv;
  p[t + 256] = v2 * inv; p[t + 384] = v3 * inv;
}

// ---------------------------------------------------------------------------
// pair_feat[n,h,c] = sum_m attn[h,n,m] * z[n,m,c] — per-n GEMM [12x512]@[512x128]
// rows padded to 16 (clamped A loads, masked D stores). 4 waves / block own
// 4 consecutive 16-wide c tiles. Second full z streaming pass -> prefetch.
// ---------------------------------------------------------------------------
__global__ __launch_bounds__(128)
void pair_feat_kernel(const float* __restrict__ attn, const float* __restrict__ z,
                      float* __restrict__ feats) {
  const int wave = threadIdx.x >> 5;
  const int lane = threadIdx.x & 31;
  const int hi = lane >> 4, l16 = lane & 15;
  const int n = blockIdx.x;
  const int c = (blockIdx.y * 4 + wave) * 16 + l16;     // < 128 always
  const int ha = l16 < NH ? l16 : NH - 1;               // clamped A row
  const float* Arow = attn + (long)ha * NN + (long)n * NRES + hi * 2;
  const float* Bcol = z + ((long)n * NRES + hi * 2) * CZ + c;
  v8f acc = {};
  for (int k = 0; k < NRES; k += 4) {
    v2f a, b;
    a.x = Arow[k];
    a.y = Arow[k + 1];
    b.x = Bcol[(long)k * CZ];
    b.y = Bcol[(long)(k + 1) * CZ];
    __builtin_prefetch(Bcol + (long)(k + 16) * CZ, 0, 0);  // next z rows
    acc = wmma4(a, b, acc);
  }
#pragma unroll
  for (int r = 0; r < 8; ++r) {
    const int h = r + hi * 8;
    if (h < NH) feats[(long)n * ODIM + h * 176 + 48 + c] = acc[r];
  }
}

// global -> local back-transform (R^T * (g - t)) + norms, into feats
__global__ __launch_bounds__(256)
void finish_pts_kernel(const float* __restrict__ ptsg, const float* __restrict__ rot,
                       const float* __restrict__ trans, float* __restrict__ feats) {
  const int idx = blockIdx.x * 256 + threadIdx.x;
  if (idx >= NRES * NH * PV) return;
  const int n = idx / (NH * PV);
  const int rem = idx - n * (NH * PV);
  const int h = rem / PV, p = rem - h * PV;
  const float* g = ptsg + (long)h * NRES * PV * 3 + n * (PV * 3) + p * 3;
  const float dx = g[0] - trans[n * 3 + 0];
  const float dy = g[1] - trans[n * 3 + 1];
  const float dz = g[2] - trans[n * 3 + 2];
  const float* R = rot + n * 9;
  const float lx = R[0] * dx + R[3] * dy + R[6] * dz;   // R^T * d
  const float ly = R[1] * dx + R[4] * dy + R[7] * dz;
  const float lz = R[2] * dx + R[5] * dy + R[8] * dz;
  float* fp = feats + (long)n * ODIM + h * 176;
  fp[16 + p * 3 + 0] = lx;
  fp[16 + p * 3 + 1] = ly;
  fp[16 + p * 3 + 2] = lz;
  fp[40 + p] = sqrtf(lx * lx + ly * ly + lz * lz);
}

// ---------------------------------------------------------------------------
extern "C" void kernel_launch(void* const* d_in, const int* in_sizes, int n_in,
                              void* d_out, int out_size, void* d_ws, size_t ws_size,
                              hipStream_t stream) {
  (void)in_sizes; (void)n_in; (void)out_size; (void)ws_size;
  const float* s    = (const float*)d_in[0];
  const float* z    = (const float*)d_in[1];
  const float* tr   = (const float*)d_in[2];
  const float* rot  = (const float*)d_in[3];
  const float* Wq   = (const float*)d_in[4];  const float* bq   = (const float*)d_in[5];
  const float* Wk   = (const float*)d_in[6];  const float* bk   = (const float*)d_in[7];
  const float* Wv   = (const float*)d_in[8];  const float* bv   = (const float*)d_in[9];
  const float* Wqp  = (const float*)d_in[10]; const float* bqp  = (const float*)d_in[11];
  const float* Wkp  = (const float*)d_in[12]; const float* bkp  = (const float*)d_in[13];
  const float* Wvp  = (const float*)d_in[14]; const float* bvp  = (const float*)d_in[15];
  const float* Wb   = (const float*)d_in[16]; const float* bb   = (const float*)d_in[17];
  const float* emb  = (const float*)d_in[18];
  const float* sl   = (const float*)d_in[19];
  const float* hw   = (const float*)d_in[20];
  const float* Wout = (const float*)d_in[21]; const float* bout = (const float*)d_in[22];
  float* out = (float*)d_out;

  // workspace layout (floats), ~19.9 MB total
  float* ws    = (float*)d_ws;
  float* q     = ws;                 // 512*192
  float* kb    = q  + 98304;         // 512*192
  float* v     = kb + 98304;         // 512*192
  float* qp    = v  + 98304;         // 512*144 (becomes q_g in place)
  float* kp    = qp + 73728;         // 512*144 (becomes k_g)
  float* vp    = kp + 73728;         // 512*288 (becomes v_g)
  float* L     = vp + 147456;        // 12*512*512 logits / attn
  float* ptsg  = L  + 3145728;       // 12*512*24 attended global points
  float* feats = ptsg + 147456;      // 512*2112

  const dim3 w4(128);                // 4 waves / block, 4 M-tiles
  // projections (M=512 -> grid.x = 8); all Ncols multiples of 16 -> FULL
  gemm_f32<true><<<dim3(8, 12), w4, 0, stream>>>(s, CS, Wq, 192, bq, q, 192, 192, CS);
  gemm_f32<true><<<dim3(8, 12), w4, 0, stream>>>(s, CS, Wk, 192, bk, kb, 192, 192, CS);
  gemm_f32<true><<<dim3(8, 12), w4, 0, stream>>>(s, CS, Wv, 192, bv, v, 192, 192, CS);
  gemm_f32<true><<<dim3(8, 9),  w4, 0, stream>>>(s, CS, Wqp, 144, bqp, qp, 144, 144, CS);
  gemm_f32<true><<<dim3(8, 9),  w4, 0, stream>>>(s, CS, Wkp, 144, bkp, kp, 144, 144, CS);
  gemm_f32<true><<<dim3(8, 18), w4, 0, stream>>>(s, CS, Wvp, 288, bvp, vp, 288, 288, CS);
  // local -> global frames (in place)
  to_global_kernel<<<(NRES * 48 + 255) / 256, 256, 0, stream>>>(qp, rot, tr, 48);
  to_global_kernel<<<(NRES * 48 + 255) / 256, 256, 0, stream>>>(kp, rot, tr, 48);
  to_global_kernel<<<(NRES * 96 + 255) / 256, 256, 0, stream>>>(vp, rot, tr, 96);
  // pair bias (z streaming GEMM) into L[h,n,m]  (NN/16 tiles / 4 waves)
  pair_bias_kernel<<<dim3(NN / 64), w4, 0, stream>>>(z, Wb, bb, L);
  // remaining logit terms
  logits_kernel<<<dim3(32, 32), dim3(384), 0, stream>>>(q, kb, qp, kp, tr, emb, sl, hw, L);
  // softmax over m
  softmax_kernel<<<dim3(NH * NRES), dim3(128), 0, stream>>>(L);
  // attn @ v (N=16 -> FULL)  and  attn @ v_g (N=24 -> masked) per head
  for (int h = 0; h < NH; ++h) {
    gemm_f32<true><<<dim3(8, 1), w4, 0, stream>>>(L + (long)h * NN, NRES, v + h * CH,
                                                  192, nullptr, feats + h * 176, ODIM,
                                                  CH, NRES);
    gemm_f32<false><<<dim3(8, 2), w4, 0, stream>>>(L + (long)h * NN, NRES, vp + h * 24,
                                                   288, nullptr,
                                                   ptsg + (long)h * NRES * 24, 24,
                                                   24, NRES);
  }
  // attn-weighted z features (second z streaming op): 8 c-tiles / 4 waves
  pair_feat_kernel<<<dim3(NRES, 2), w4, 0, stream>>>(L, z, feats);
  // back-transform points + norms
  finish_pts_kernel<<<(NRES * NH * PV + 255) / 256, 256, 0, stream>>>(ptsg, rot, tr, feats);
  // final projection (N=384 -> FULL)
  gemm_f32<true><<<dim3(8, 24), w4, 0, stream>>>(feats, ODIM, Wout, CS, bout, out, CS,
                                                 CS, ODIM);
}